// CustomLoss_91001767068026
// MI455X (gfx1250) — compile-verified
//
#include <hip/hip_runtime.h>

// ---------------------------------------------------------------------------
// CDNA5 / gfx1250 implementation.
// Heavy math runs through V_WMMA_F32_16X16X4_F32 (fp32 to match reference).
// ---------------------------------------------------------------------------

typedef float v2f __attribute__((ext_vector_type(2)));
typedef float v8f __attribute__((ext_vector_type(8)));

#define NCOMP  30
#define CUTOFF 5.0f

__device__ __forceinline__ v8f wmma_f32_16x16x4(v2f a, v2f b, v8f c) {
  // D = A(16x4 f32) * B(4x16 f32) + C(16x16 f32)
  // args: (neg_a, A, neg_b, B, c_mod, C, reuse_a, reuse_b)
  return __builtin_amdgcn_wmma_f32_16x16x4_f32(
      false, a, false, b, (short)0, c, false, false);
}

// ---------------------------------------------------------------------------
// Kernel 1+2: flat-index -> breast slot map (for the .at[breast_idx].set scatter)
// ---------------------------------------------------------------------------
__global__ void k_init_map(int* __restrict__ map, int n) {
  int i = blockIdx.x * blockDim.x + threadIdx.x;
  if (i < n) map[i] = -1;
}

__global__ void k_scatter_map(int* __restrict__ map,
                              const int* __restrict__ bidx, int nb) {
  int i = blockIdx.x * blockDim.x + threadIdx.x;
  if (i < nb) {
    int g = 3 * bidx[i];
    map[g + 0] = i;
    map[g + 1] = i;
    map[g + 2] = i;
  }
}

// ---------------------------------------------------------------------------
// Kernel 3: gathered skinny GEMM  base[g] = dot(U[g,:], x_b) + mean[g]
// Rows (gathered flat indices) come from three regions:
//   [0, 3*NB)          -> 3*breast_idx[r/3] + r%3   -> dotB[b][r]
//   [3*NB, 3*NB+3*NO)  -> 3*outline_idx[.] + .      -> dotO[b][.]
//   [.., +L)           -> line_idx[.]               -> dotL[b][.]
// One wave: M=16 gathered rows, N=16 sample columns (8 valid), K=32 (30 pad 0),
// 8 accumulating v_wmma_f32_16x16x4_f32 steps. The padded B matrix (X^T) is
// staged zero-filled in LDS so the inner loop has NO guarded loads.
// ---------------------------------------------------------------------------
__device__ __forceinline__ int row_to_flat(int row, int R1, int R2,
                                           const int* __restrict__ bidx,
                                           const int* __restrict__ oidx,
                                           const int* __restrict__ lidx) {
  if (row < R1) return 3 * bidx[row / 3] + (row % 3);
  row -= R1;
  if (row < R2 - R1) return 3 * oidx[row / 3] + (row % 3);
  row -= (R2 - R1);
  return lidx[row];
}

__global__ __launch_bounds__(256)
void k_gemm_gather(const float* __restrict__ U, const float* __restrict__ X,
                   const float* __restrict__ mean,
                   const int* __restrict__ bidx, const int* __restrict__ oidx,
                   const int* __restrict__ lidx,
                   float* __restrict__ dotB, float* __restrict__ dotO,
                   float* __restrict__ dotL,
                   int NB, int NO, int L, int B) {
  const int R1 = 3 * NB;
  const int R2 = R1 + 3 * NO;
  const int R  = R2 + L;

  // --- stage zero-padded B matrix: sX[sample n][K pair] = (x[n,2kp], x[n,2kp+1])
  __shared__ v2f sX[16][16];  // 16 samples x 32 K values, zero-padded
  {
    int tid = threadIdx.x;            // 256 threads -> one v2f each
    int n   = tid >> 4;               // 0..15
    int kp  = tid & 15;               // pair index, k = 2*kp
    int k   = 2 * kp;
    // clamped, always-valid addresses + value select (no divergent-load branch)
    int na = min(n, B - 1);
    int ka = min(k, NCOMP - 2);
    float x0 = X[na * NCOMP + ka + 0];
    float x1 = X[na * NCOMP + ka + 1];
    v2f t;
    t.x = (n < B && k + 0 < NCOMP) ? x0 : 0.0f;
    t.y = (n < B && k + 1 < NCOMP) ? x1 : 0.0f;
    sX[n][kp] = t;
  }
  __syncthreads();

  int wave = (blockIdx.x * blockDim.x + threadIdx.x) >> 5;  // global wave id
  int lane = threadIdx.x & 31;
  int m    = lane & 15;
  int row0 = wave * 16;
  if (row0 >= R) return;  // wave-uniform: EXEC stays all-ones for WMMA

  // A operand: lane<16 holds (K0,K1) of row m; lane>=16 holds (K2,K3) of row m.
  int row = row0 + m;
  int gA  = row_to_flat(min(row, R - 1), R1, R2, bidx, oidx, lidx);
  const float* urow = U + (long)gA * NCOMP;

  v8f acc = {};
#pragma unroll
  for (int kt = 0; kt < 8; ++kt) {
    int k0 = 4 * kt + ((lane < 16) ? 0 : 2);
    // unconditional aligned v2f load from a clamped (valid) address,
    // then zero-select the out-of-range tail (kt==7, lanes>=16 only)
    int kA = (k0 < NCOMP) ? k0 : 0;
    v2f av = *(const v2f*)(urow + kA);          // (g*30+k0) even -> 8B aligned
    av.x = (k0 < NCOMP) ? av.x : 0.0f;
    av.y = (k0 < NCOMP) ? av.y : 0.0f;
    v2f bv = sX[m][k0 >> 1];                    // padded: no guard needed
    acc = wmma_f32_16x16x4(av, bv, acc);
  }

  // C layout: VGPR r = rows {r, r+8} in lane halves, column = lane&15 = sample.
  if (m < B) {
#pragma unroll
    for (int r = 0; r < 8; ++r) {
      int wrow = row0 + r + ((lane < 16) ? 0 : 8);
      if (wrow < R) {
        int g   = row_to_flat(wrow, R1, R2, bidx, oidx, lidx);
        float v = acc[r] + mean[g];
        if (wrow < R1)      dotB[(long)m * R1 + wrow] = v;
        else if (wrow < R2) dotO[(long)m * 3 * NO + (wrow - R1)] = v;
        else                dotL[(long)m * L + (wrow - R2)] = v;
      }
    }
  }
}

// ---------------------------------------------------------------------------
// Kernel 4: per-breast-point min distance to outline + cutoff weight.
// d2(i,j) = |bp_i|^2 + |op_j|^2 - 2*dot(bp_i, op_j); the Gram part is a K=3
// GEMM done with v_wmma_f32_16x16x4_f32. Outline points (+norm) staged in LDS.
// grid = (NB/16/8, B), block = 256 (8 waves, each owns 16 breast rows).
// ---------------------------------------------------------------------------
__global__ __launch_bounds__(256)
void k_min_dist_w(const float* __restrict__ dotB, const float* __restrict__ dotO,
                  float* __restrict__ wArr, int NB, int NO) {
  __shared__ float4 sO[1024];  // x, y, z, |p|^2

  int b   = blockIdx.y;
  int tid = threadIdx.x;

  const float* ob = dotO + (long)b * 3 * NO;
  for (int j = tid; j < NO; j += blockDim.x) {
    float x = ob[3 * j + 0], y = ob[3 * j + 1], z = ob[3 * j + 2];
    sO[j] = make_float4(x, y, z, x * x + y * y + z * z);
  }
  __syncthreads();

  int wid = tid >> 5, lane = tid & 31, m = lane & 31 & 15;
  m = lane & 15;
  int rowTile = blockIdx.x * (blockDim.x >> 5) + wid;
  int row     = rowTile * 16 + m;

  const float* bb = dotB + (long)b * 3 * NB;
  float px = bb[3 * row + 0], py = bb[3 * row + 1], pz = bb[3 * row + 2];
  float bn = px * px + py * py + pz * pz;

  v2f a;  // lane<16: (x,y) K={0,1}; lane>=16: (z,0) K={2,3} (K=3 padded to 4)
  a.x = (lane < 16) ? px : pz;
  a.y = (lane < 16) ? py : 0.0f;

  float mn[8];
#pragma unroll
  for (int r = 0; r < 8; ++r) mn[r] = 3.0e38f;

  int jtiles = NO >> 4;
#pragma unroll 4
  for (int jt = 0; jt < jtiles; ++jt) {
    float4 p = sO[jt * 16 + m];
    v2f bv;
    bv.x = (lane < 16) ? p.x : p.z;
    bv.y = (lane < 16) ? p.y : 0.0f;
    v8f c = {};
    c = wmma_f32_16x16x4(a, bv, c);  // c[r] = dot(bp_row, op_col)
    float on = p.w;
#pragma unroll
    for (int r = 0; r < 8; ++r)
      mn[r] = fminf(mn[r], fmaf(-2.0f, c[r], on));
  }

  // min over the 16 columns held by each lane half (halves are independent rows)
#pragma unroll
  for (int mask = 1; mask <= 8; mask <<= 1) {
#pragma unroll
    for (int r = 0; r < 8; ++r)
      mn[r] = fminf(mn[r], __shfl_xor(mn[r], mask, 32));
  }

  float* wb = wArr + (long)b * NB + rowTile * 16;
#pragma unroll
  for (int r = 0; r < 8; ++r) {
    float bnlo = __shfl(bn, r, 32);       // |bp|^2 of row r
    float bnhi = __shfl(bn, 8 + r, 32);   // |bp|^2 of row 8+r
    if (lane == 0) {
      float d = sqrtf(fmaxf(bnlo + mn[r], 0.0f));
      wb[r] = (d > CUTOFF) ? 1.0f : d / CUTOFF;
    }
    if (lane == 16) {
      float d = sqrtf(fmaxf(bnhi + mn[r], 0.0f));
      wb[8 + r] = (d > CUTOFF) ? 1.0f : d / CUTOFF;
    }
  }
}

// ---------------------------------------------------------------------------
// Kernel 5: loss = sum_t sqrt( sum_j (model[tri[t,j]] - target[b,3t])^2 )
// model[g] = blended (if g hits a breast coordinate, via slot map) else base.
// One block per sample; wave shfl + LDS reduction.
// ---------------------------------------------------------------------------
__global__ __launch_bounds__(256)
void k_final_loss(const float* __restrict__ dotB, const float* __restrict__ dotL,
                  const float* __restrict__ wArr, const int* __restrict__ map,
                  const int* __restrict__ lidx, const float* __restrict__ target,
                  float* __restrict__ out, int NB, int L) {
  __shared__ float red[8];
  int b = blockIdx.x;
  int tid = threadIdx.x;
  int ntri = L / 3;

  const float* dB = dotB + (long)b * 3 * NB;
  const float* dL = dotL + (long)b * L;
  const float* wb = wArr + (long)b * NB;
  const float* tg = target + (long)b * L;

  float local = 0.0f;
  for (int t = tid; t < ntri; t += blockDim.x) {
    float tj = tg[3 * t];
    float s = 0.0f;
#pragma unroll
    for (int j = 0; j < 3; ++j) {
      int e = 3 * t + j;
      int g = lidx[e];
      int slot = map[g];
      float val;
      if (slot >= 0) {
        float v  = dB[3 * slot + (g % 3)];  // adjusted == base point coord
        float wv = wb[slot];
        val = wv * v + (1.0f - wv) * v;     // faithful blend (≈ v)
      } else {
        val = dL[e];
      }
      float d = val - tj;
      s += d * d;
    }
    local += sqrtf(s);
  }

#pragma unroll
  for (int mask = 16; mask >= 1; mask >>= 1)
    local += __shfl_xor(local, mask, 32);
  if ((tid & 31) == 0) red[tid >> 5] = local;
  __syncthreads();
  if (tid == 0) {
    float s = 0.0f;
#pragma unroll
    for (int i = 0; i < 8; ++i) s += red[i];
    out[b] = s;
  }
}

// ---------------------------------------------------------------------------
extern "C" void kernel_launch(void* const* d_in, const int* in_sizes, int n_in,
                              void* d_out, int out_size, void* d_ws, size_t ws_size,
                              hipStream_t stream) {
  (void)n_in; (void)out_size; (void)ws_size;
  const float* X      = (const float*)d_in[0];  // (B, 30)
  const float* target = (const float*)d_in[1];  // (B, L)
  const float* U      = (const float*)d_in[2];  // (3V, 30)
  const float* mean   = (const float*)d_in[3];  // (3V,)
  const int*   bidx   = (const int*)d_in[4];    // (NB,)
  const int*   oidx   = (const int*)d_in[5];    // (NO,)
  const int*   lidx   = (const int*)d_in[6];    // (L,)

  const int B  = in_sizes[0] / NCOMP;  // 8
  const int V3 = in_sizes[3];          // 300000
  const int NB = in_sizes[4];          // 8192
  const int NO = in_sizes[5];          // 1024
  const int L  = in_sizes[6];          // 3072

  // Workspace layout (~2.5 MB total)
  char* ws = (char*)d_ws;
  size_t off = 0;
  auto alignup = [](size_t x) { return (x + 255) & ~(size_t)255; };
  int*   map  = (int*)(ws + off);   off = alignup(off + (size_t)V3 * 4);
  float* dotB = (float*)(ws + off); off = alignup(off + (size_t)B * 3 * NB * 4);
  float* dotO = (float*)(ws + off); off = alignup(off + (size_t)B * 3 * NO * 4);
  float* dotL = (float*)(ws + off); off = alignup(off + (size_t)B * L * 4);
  float* wArr = (float*)(ws + off); off = alignup(off + (size_t)B * NB * 4);

  k_init_map<<<(V3 + 255) / 256, 256, 0, stream>>>(map, V3);
  k_scatter_map<<<(NB + 255) / 256, 256, 0, stream>>>(map, bidx, NB);

  const int R = 3 * NB + 3 * NO + L;          // 30720 gathered rows
  const int tiles = (R + 15) / 16;            // 1920 wave tiles
  k_gemm_gather<<<(tiles + 7) / 8, 256, 0, stream>>>(
      U, X, mean, bidx, oidx, lidx, dotB, dotO, dotL, NB, NO, L, B);

  dim3 gC((NB / 16) / 8, B);                  // (64, 8)
  k_min_dist_w<<<gC, 256, 0, stream>>>(dotB, dotO, wArr, NB, NO);

  k_final_loss<<<B, 256, 0, stream>>>(dotB, dotL, wArr, map, lidx, target,
                                      (float*)d_out, NB, L);
}